// spuLayerTransformer_23785528885334
// MI455X (gfx1250) — compile-verified
//
#include <hip/hip_runtime.h>

// Elementwise SPU + interval-box bounds over fp32 vectors.
// Memory-bound: ~402 MB streamed once -> ~17.3 us floor at 23.3 TB/s HBM.
// Strategy: b128 vector loads/stores with non-temporal (streaming) cache
// hints, wave32 blocks of 256 threads (8 waves), branch-free select math.

typedef __attribute__((ext_vector_type(4))) float v4f;

__device__ __forceinline__ float spu_scalar(float v) {
    // x >= 0 : x*x - 0.5
    // x <  0 : sigmoid(-x) - 1 = 1/(1 + exp(x)) - 1
    float pos = __builtin_fmaf(v, v, -0.5f);
    float e   = __expf(v);                 // v_exp_f32 (x<0 -> e in (0,1))
    float neg = __frcp_rn(1.0f + e) - 1.0f; // v_rcp_f32; x>>0 -> 0-1=-1, discarded by select
    return (v >= 0.0f) ? pos : neg;
}

__global__ void __launch_bounds__(256)
spu_box_vec4_kernel(const float* __restrict__ x,
                    const float* __restrict__ l,
                    const float* __restrict__ u,
                    float* __restrict__ x_out,
                    float* __restrict__ l_out,
                    float* __restrict__ u_out,
                    int n4) {
    int i = blockIdx.x * blockDim.x + threadIdx.x;
    if (i >= n4) return;

    // 128-bit non-temporal streaming loads (global_load_b128, TH=NT)
    v4f xv = __builtin_nontemporal_load((const v4f*)x + i);
    v4f lv = __builtin_nontemporal_load((const v4f*)l + i);
    v4f uv = __builtin_nontemporal_load((const v4f*)u + i);

    v4f xo, lo, uo;
#pragma unroll
    for (int k = 0; k < 4; ++k) {
        float xs = xv[k];
        float ls = lv[k];
        float us = uv[k];

        float sx = spu_scalar(xs);
        float sl = spu_scalar(ls);
        float su = spu_scalar(us);

        // u_out = (u <= 0) ? spu(l) : spu(u)
        float uo_s = (us <= 0.0f) ? sl : su;
        // l_out = (l >= 0) ? spu(l) : ((u <= 0) ? spu(u) : -0.5)
        float lo_s = (ls >= 0.0f) ? sl : ((us <= 0.0f) ? su : -0.5f);

        xo[k] = sx;
        lo[k] = lo_s;
        uo[k] = uo_s;
    }

    // 128-bit non-temporal streaming stores (global_store_b128, TH=NT)
    __builtin_nontemporal_store(xo, (v4f*)x_out + i);
    __builtin_nontemporal_store(lo, (v4f*)l_out + i);
    __builtin_nontemporal_store(uo, (v4f*)u_out + i);
}

// Scalar tail (only launched when n % 4 != 0; never for DIM = 2^24).
__global__ void spu_box_tail_kernel(const float* __restrict__ x,
                                    const float* __restrict__ l,
                                    const float* __restrict__ u,
                                    float* __restrict__ x_out,
                                    float* __restrict__ l_out,
                                    float* __restrict__ u_out,
                                    int start, int n) {
    int i = start + blockIdx.x * blockDim.x + threadIdx.x;
    if (i >= n) return;
    float xs = x[i], ls = l[i], us = u[i];
    float sx = spu_scalar(xs);
    float sl = spu_scalar(ls);
    float su = spu_scalar(us);
    x_out[i] = sx;
    u_out[i] = (us <= 0.0f) ? sl : su;
    l_out[i] = (ls >= 0.0f) ? sl : ((us <= 0.0f) ? su : -0.5f);
}

extern "C" void kernel_launch(void* const* d_in, const int* in_sizes, int n_in,
                              void* d_out, int out_size, void* d_ws, size_t ws_size,
                              hipStream_t stream) {
    (void)n_in; (void)out_size; (void)d_ws; (void)ws_size;

    const float* x = (const float*)d_in[0];
    const float* l = (const float*)d_in[1];
    const float* u = (const float*)d_in[2];

    const int n = in_sizes[0];           // DIM = 16777216
    float* out   = (float*)d_out;        // tuple concat: x_out | l_out | u_out
    float* x_out = out;
    float* l_out = out + (size_t)n;
    float* u_out = out + 2 * (size_t)n;

    const int n4   = n >> 2;             // float4 packets
    const int tail = n & 3;

    const int block = 256;               // 8 wave32 waves per block
    if (n4 > 0) {
        const int grid = (n4 + block - 1) / block;
        spu_box_vec4_kernel<<<grid, block, 0, stream>>>(x, l, u,
                                                        x_out, l_out, u_out, n4);
    }
    if (tail > 0) {
        spu_box_tail_kernel<<<1, block, 0, stream>>>(x, l, u,
                                                     x_out, l_out, u_out,
                                                     n4 << 2, n);
    }
}